// GCN_8_72782515798116
// MI455X (gfx1250) — compile-verified
//
#include <hip/hip_runtime.h>
#include <hip/hip_bf16.h>
#include <math.h>

// ---------------- problem constants (from reference) ----------------
#define N_NODES   24
#define F_IN      1048576
#define HID       8
#define N_EDGES   240
#define FLAT_DIM  (N_NODES * HID)   // 192
#define FC1_DIM   128
#define N_CLASSES 2

// ---------------- GEMM partition ----------------
#define NBLK    512                  // blocks for the big reduction
#define WPB     8                    // waves per block (256 threads)
#define F_CHUNK (F_IN / NBLK)        // 2048 features per block
#define F_WAVE  (F_CHUNK / WPB)      // 256 features per wave
#define KSTEPS  (F_WAVE / 4)         // 64 WMMA k-steps per wave per tile

typedef __attribute__((ext_vector_type(2))) float v2f;
typedef __attribute__((ext_vector_type(8))) float v8f;

// Kernel 1: partial of h_pre = x[24,F] @ w[F,8] over this block's f-chunk,
// via V_WMMA_F32_16X16X4_F32. HBM-bound (132 MB total, ~5.7us roofline);
// the WMMA math is free. Hot loop is branch-free: pad lanes load clamped
// (valid, cache-aliased) addresses and are zeroed with cndmask selects.
__global__ void __launch_bounds__(256)
gcn_gemm_wmma(const float* __restrict__ x,     // [24, F_IN] row-major
              const float* __restrict__ w,     // [F_IN, 8]  row-major
              float* __restrict__ partial)     // [NBLK, 512]
{
    const int tid  = threadIdx.x;
    const int lane = tid & 31;
    const int wave = tid >> 5;
    const int blk  = blockIdx.x;

    // A 16x4 f32 layout: lanes 0-15 hold M=lane with K0,K1; lanes 16-31 hold
    // the same M=lane-16 with K2,K3. B mirrors with N=lane&15.
    const int  m      = lane & 15;              // M (tile row) / N (tile col)
    const int  koff   = (lane >= 16) ? 2 : 0;   // K pair owned by this half-wave
    const bool row1ok = (m < 8);                // tile1 node 16+m valid only m<8
    const bool colok  = (m < 8);                // only 8 real output columns

    const int f0 = blk * F_CHUNK + wave * F_WAVE;

    // Per-lane base pointers (computed once; loop uses constant offsets).
    const float* __restrict__ pa0 = x + (size_t)m * F_IN + f0 + koff;              // nodes 0..15
    const float* __restrict__ pa1 = x + (size_t)(16 + (m & 7)) * F_IN + f0 + koff; // nodes 16..23 (clamped)
    const float* __restrict__ pb  = w + (size_t)(f0 + koff) * HID + (m & 7);       // w column (clamped)

    v8f c0 = {0.f,0.f,0.f,0.f,0.f,0.f,0.f,0.f};
    v8f c1 = {0.f,0.f,0.f,0.f,0.f,0.f,0.f,0.f};

    #pragma unroll 4
    for (int s = 0; s < KSTEPS; ++s) {
        // Unconditional aligned loads (8B for x K-pairs, 2x4B for w).
        const float2 xa0 = *(const float2*)(pa0 + s * 4);
        const float2 xa1 = *(const float2*)(pa1 + s * 4);
        const float  bx  = pb[s * 32];          // w[fk  ][col]
        const float  by  = pb[s * 32 + 8];      // w[fk+1][col]

        v2f a0; a0.x = xa0.x;                 a0.y = xa0.y;
        v2f a1; a1.x = row1ok ? xa1.x : 0.f;  a1.y = row1ok ? xa1.y : 0.f;
        v2f b;  b.x  = colok  ? bx    : 0.f;  b.y  = colok  ? by    : 0.f;

        c0 = __builtin_amdgcn_wmma_f32_16x16x4_f32(false, a0, false, b,
                                                   (short)0, c0, false, false);
        c1 = __builtin_amdgcn_wmma_f32_16x16x4_f32(false, a1, false, b,
                                                   (short)0, c1, false, false);
    }

    // Deterministic 8-wave block reduction through LDS.
    __shared__ float red[WPB][512];             // 16 KB
    #pragma unroll
    for (int v = 0; v < 8; ++v) {
        red[wave][      lane * 8 + v] = c0[v];
        red[wave][256 + lane * 8 + v] = c1[v];
    }
    __syncthreads();
    for (int s = tid; s < 512; s += 256) {
        float acc = 0.f;
        #pragma unroll
        for (int wv = 0; wv < WPB; ++wv) acc += red[wv][s];
        partial[(size_t)blk * 512 + s] = acc;
    }
}

// Kernel 2: fixed-order partial reduction + whole GCN tail (tiny).
__global__ void __launch_bounds__(256)
gcn_tail(const float* __restrict__ partial,      // [NBLK, 512]
         const int*   __restrict__ ei,           // [2, 240]
         const float* __restrict__ conv_b,       // [8]
         const float* __restrict__ fc1_w,        // [128, 192]
         const float* __restrict__ fc1_b,        // [128]
         const float* __restrict__ fc2_w,        // [2, 128]
         const float* __restrict__ fc2_b,        // [2]
         float*       __restrict__ out)          // [2]
{
    __shared__ float hpre[N_NODES][HID];          // x @ W (pre-norm)
    __shared__ float aggs[N_NODES][HID];
    __shared__ float hact[FLAT_DIM];              // relu(agg + b), node-major
    __shared__ float dis[N_NODES];
    __shared__ float z[FC1_DIM];

    const int tid = threadIdx.x;

    // 1) reduce NBLK partials per C-tile slot, fixed order -> deterministic
    for (int s = tid; s < 512; s += 256) {
        float acc = 0.f;
        for (int b = 0; b < NBLK; ++b) acc += partial[(size_t)b * 512 + s];
        const int t    = s >> 8;                  // tile
        const int l    = (s >> 3) & 31;           // lane
        const int v    = s & 7;                   // C vgpr
        const int node = t * 16 + ((l < 16) ? v : v + 8);
        const int c    = l & 15;
        if (c < HID && node < N_NODES) hpre[node][c] = acc;
    }
    __syncthreads();

    // 2) degrees (targets + self loops), dis = rsqrt(deg)
    if (tid == 0) {
        for (int n = 0; n < N_NODES; ++n) dis[n] = 1.0f;   // self loop
        for (int e = 0; e < N_EDGES; ++e) dis[ei[N_EDGES + e]] += 1.0f;
        for (int n = 0; n < N_NODES; ++n) dis[n] = rsqrtf(dis[n]);
    }
    __syncthreads();

    // 3) symmetric-norm scatter-add; thread c owns output column c (race-free)
    if (tid < HID) {
        const int c = tid;
        for (int n = 0; n < N_NODES; ++n)
            aggs[n][c] = hpre[n][c] * dis[n] * dis[n];     // self-loop term
        for (int e = 0; e < N_EDGES; ++e) {
            const int r = ei[e], t = ei[N_EDGES + e];
            aggs[t][c] += hpre[r][c] * dis[r] * dis[t];
        }
        const float bc = conv_b[c];
        for (int n = 0; n < N_NODES; ++n) {
            const float hv = aggs[n][c] + bc;
            hact[n * HID + c] = hv > 0.f ? hv : 0.f;       // relu, node-major flat
        }
    }
    __syncthreads();

    // 4) fc1: z = fc1_w @ flat + fc1_b
    if (tid < FC1_DIM) {
        float acc = fc1_b[tid];
        for (int i = 0; i < FLAT_DIM; ++i)
            acc = fmaf(fc1_w[tid * FLAT_DIM + i], hact[i], acc);
        z[tid] = acc;
    }
    __syncthreads();

    // 5) fc2 + log_softmax
    if (tid == 0) {
        float l0 = fc2_b[0], l1 = fc2_b[1];
        for (int j = 0; j < FC1_DIM; ++j) {
            l0 = fmaf(fc2_w[j],           z[j], l0);
            l1 = fmaf(fc2_w[FC1_DIM + j], z[j], l1);
        }
        const float mx  = fmaxf(l0, l1);
        const float lse = mx + logf(expf(l0 - mx) + expf(l1 - mx));
        out[0] = l0 - lse;
        out[1] = l1 - lse;
    }
}

extern "C" void kernel_launch(void* const* d_in, const int* in_sizes, int n_in,
                              void* d_out, int out_size, void* d_ws, size_t ws_size,
                              hipStream_t stream) {
    const float* x      = (const float*)d_in[0];
    const int*   ei     = (const int*)  d_in[1];
    const float* conv_w = (const float*)d_in[2];
    const float* conv_b = (const float*)d_in[3];
    const float* fc1_w  = (const float*)d_in[4];
    const float* fc1_b  = (const float*)d_in[5];
    const float* fc2_w  = (const float*)d_in[6];
    const float* fc2_b  = (const float*)d_in[7];
    float* out = (float*)d_out;
    float* partial = (float*)d_ws;               // NBLK*512 floats = 1 MB

    gcn_gemm_wmma<<<NBLK, 256, 0, stream>>>(x, conv_w, partial);
    gcn_tail<<<1, 256, 0, stream>>>(partial, ei, conv_b, fc1_w, fc1_b,
                                    fc2_w, fc2_b, out);
}